// SGCN_48550310314177
// MI455X (gfx1250) — compile-verified
//
#include <hip/hip_runtime.h>
#include <hip/hip_bf16.h>

// ---------------------------------------------------------------------------
// SGCN: h = einsum(x^T, W) via V_WMMA_F32_16X16X4_F32; edge scatter with
// global_atomic_add_f32; 2-pass BatchNorm + ReLU.
// Shapes: B=4, C_in=C_out=32, N=10000, T=12, E=160000.
// h/agg layout: [B, N, C_out*T] -> 1536 contiguous bytes per (b,node).
// ---------------------------------------------------------------------------

typedef __attribute__((ext_vector_type(2))) float v2f;
typedef __attribute__((ext_vector_type(8))) float v8f;

#define SG_B    4
#define SG_CIN  32
#define SG_COUT 32
#define SG_N    10000
#define SG_T    12
#define SG_E    160000
#define SG_NT   (SG_N * SG_T)            // 120000 rows per batch
#define SG_ROW  (SG_COUT * SG_T)         // 384 floats per (b,node)
#define SG_HSZ  (SG_B * SG_N * SG_ROW)   // 15,360,000 floats
#define SG_MT   (SG_NT / 16)             // 7500 M-tiles per batch

// --------------------------- GEMM via WMMA f32 -----------------------------
// One wave handles a 16-row (n,t) tile x all 32 output channels.
// A: rows r=(n*T+t), cols c (stride NT in memory -> column major, loads
// directly into the 16x4 A fragment). B: W[c,d] row-major.
__global__ __launch_bounds__(128) void sgcn_gemm_wmma(
    const float* __restrict__ x,   // [B, C_in, N, T]
    const float* __restrict__ W,   // [C_in, C_out]
    float* __restrict__ h)         // [B, N, C_out, T]
{
    int wave = (blockIdx.x * blockDim.x + threadIdx.x) >> 5;  // 0..B*SG_MT-1
    if (wave >= SG_B * SG_MT) return;
    int bb   = wave / SG_MT;
    int mt   = wave % SG_MT;
    int lane = threadIdx.x & 31;
    int half = lane >> 4;    // which half-wave
    int l16  = lane & 15;
    int row0 = mt * 16;

    const float* xb = x + (size_t)bb * SG_CIN * SG_NT;

    v8f c0 = {};  // D columns 0..15
    v8f c1 = {};  // D columns 16..31
#pragma unroll
    for (int kk = 0; kk < 8; ++kk) {
        int kbase = kk * 4 + 2 * half;   // fragment component v holds K=kbase+v
        v2f a;
        a.x = xb[(kbase + 0) * SG_NT + row0 + l16];
        a.y = xb[(kbase + 1) * SG_NT + row0 + l16];
        v2f w0, w1;
        w0.x = W[(kbase + 0) * SG_COUT + l16];
        w0.y = W[(kbase + 1) * SG_COUT + l16];
        w1.x = W[(kbase + 0) * SG_COUT + 16 + l16];
        w1.y = W[(kbase + 1) * SG_COUT + 16 + l16];
        c0 = __builtin_amdgcn_wmma_f32_16x16x4_f32(
                 false, a, false, w0, (short)0, c0, false, false);
        c1 = __builtin_amdgcn_wmma_f32_16x16x4_f32(
                 false, a, false, w1, (short)0, c1, false, false);
    }

    // D layout: VGPR v -> M = v + 8*half, N = l16 (tile0) / 16+l16 (tile1)
#pragma unroll
    for (int v = 0; v < 8; ++v) {
        int r    = row0 + v + 8 * half;
        int node = r / SG_T;
        int t    = r - node * SG_T;
        float* hp = h + ((size_t)(bb * SG_N + node)) * SG_ROW + t;
        hp[(size_t)l16 * SG_T]        = c0[v];
        hp[(size_t)(16 + l16) * SG_T] = c1[v];
    }
}

// ------------------ init agg with bias, zero BN stats ----------------------
__global__ __launch_bounds__(256) void sgcn_init(
    float* __restrict__ agg, const float* __restrict__ bias,
    float* __restrict__ stats)
{
    int idx = blockIdx.x * 256 + threadIdx.x;
    if (blockIdx.x == 0 && threadIdx.x < 128) stats[threadIdx.x] = 0.0f;
    if (idx < SG_HSZ) {
        int d = (idx / SG_T) & (SG_COUT - 1);
        agg[idx] = bias[d];
    }
}

// ------------------------------ edge scatter -------------------------------
// One thread per (b, e, j) element: 1536B contiguous per edge per batch.
__global__ __launch_bounds__(256) void sgcn_scatter(
    const float*     __restrict__ h,
    const long long* __restrict__ ei,   // [2, E] int64
    const float*     __restrict__ ew,   // [E]
    float*           __restrict__ agg)
{
    int bb  = blockIdx.y;
    int idx = blockIdx.x * 256 + threadIdx.x;   // 0 .. E*384-1 (exact)
    int j   = idx % SG_ROW;
    int e   = idx / SG_ROW;
    int src = (int)ei[e];
    int dst = (int)ei[SG_E + e];
    float w = ew[e];
    float val = h[(size_t)(bb * SG_N + src) * SG_ROW + j] * w;
    atomicAdd(&agg[(size_t)(bb * SG_N + dst) * SG_ROW + j], val);
}

// --------------- BN reduction: one block per channel, no atomics -----------
__global__ __launch_bounds__(256) void sgcn_stats(
    const float* __restrict__ agg,
    const float* __restrict__ gamma,
    const float* __restrict__ beta,
    float* __restrict__ stats)   // out: stats[d]=scale, stats[32+d]=shift
{
    int d = blockIdx.x;          // 0..31
    __shared__ float ssum[256];
    __shared__ float ssq[256];
    float s = 0.0f, q = 0.0f;
    const int ROWS = SG_B * SG_N;   // 40000
    for (int r = threadIdx.x; r < ROWS; r += 256) {
        const float* p = agg + (size_t)r * SG_ROW + d * SG_T;
#pragma unroll
        for (int t = 0; t < SG_T; ++t) { float v = p[t]; s += v; q += v * v; }
    }
    ssum[threadIdx.x] = s;
    ssq[threadIdx.x]  = q;
    __syncthreads();
    for (int off = 128; off > 0; off >>= 1) {
        if (threadIdx.x < (unsigned)off) {
            ssum[threadIdx.x] += ssum[threadIdx.x + off];
            ssq[threadIdx.x]  += ssq[threadIdx.x + off];
        }
        __syncthreads();
    }
    if (threadIdx.x == 0) {
        const float inv_cnt = 1.0f / (float)(SG_B * SG_N * SG_T);
        float mean  = ssum[0] * inv_cnt;
        float var   = ssq[0] * inv_cnt - mean * mean;
        float scale = gamma[d] * rsqrtf(var + 1e-5f);
        stats[d]            = scale;
        stats[SG_COUT + d]  = beta[d] - mean * scale;
    }
}

// --------------- finalize: normalize + ReLU + transpose to output ----------
__global__ __launch_bounds__(256) void sgcn_final(
    const float* __restrict__ agg,     // [B, N, C_out, T]
    const float* __restrict__ stats,
    float* __restrict__ out)           // [B, C_out, N, T]
{
    int idx = blockIdx.x * 256 + threadIdx.x;
    if (idx >= SG_HSZ) return;
    int t  = idx % SG_T;
    int r  = idx / SG_T;
    int n  = r % SG_N;
    int r2 = r / SG_N;
    int d  = r2 & (SG_COUT - 1);
    int bb = r2 >> 5;
    float v = agg[((size_t)(bb * SG_N + n) * SG_COUT + d) * SG_T + t];
    float y = v * stats[d] + stats[SG_COUT + d];
    out[idx] = fmaxf(y, 0.0f);
}

// ---------------------------------------------------------------------------
extern "C" void kernel_launch(void* const* d_in, const int* in_sizes, int n_in,
                              void* d_out, int out_size, void* d_ws, size_t ws_size,
                              hipStream_t stream)
{
    const float*     x     = (const float*)d_in[0];
    const long long* ei    = (const long long*)d_in[1];
    const float*     ew    = (const float*)d_in[2];
    const float*     W     = (const float*)d_in[3];
    const float*     bias  = (const float*)d_in[4];
    const float*     gamma = (const float*)d_in[5];
    const float*     beta  = (const float*)d_in[6];
    float*           out   = (float*)d_out;

    float* h     = (float*)d_ws;
    float* agg   = h + SG_HSZ;
    float* stats = agg + SG_HSZ;

    // 1) feature transform (WMMA f32): B*7500 waves, 4 waves/block
    sgcn_gemm_wmma<<<(SG_B * SG_MT + 3) / 4, 128, 0, stream>>>(x, W, h);

    // 2) agg = bias broadcast; zero stats
    sgcn_init<<<(SG_HSZ + 255) / 256, 256, 0, stream>>>(agg, bias, stats);

    // 3) weighted edge scatter-add (L2-resident gather + float atomics)
    dim3 sg_grid(SG_E * SG_ROW / 256, SG_B, 1);   // 240000 x 4
    sgcn_scatter<<<sg_grid, 256, 0, stream>>>(h, ei, ew, agg);

    // 4) per-channel BN stats -> scale/shift
    sgcn_stats<<<SG_COUT, 256, 0, stream>>>(agg, gamma, beta, stats);

    // 5) normalize + ReLU + transpose to [B, C_out, N, T]
    sgcn_final<<<(SG_HSZ + 255) / 256, 256, 0, stream>>>(agg, stats, out);
}